// Difference3DCostVolume_59081570125093
// MI455X (gfx1250) — compile-verified
//
#include <hip/hip_runtime.h>
#include <stdint.h>

#define MAX_DISP  48
#define PAD_VALUE 1.0f
#define B_   4
#define C_   32
#define H_   96
#define W_   312
#define CHUNKS (W_ / 4)          // 78 float4 chunks per row
#define BLOCK  256               // 8 wave32s

typedef float v4f __attribute__((ext_vector_type(4)));

__global__ __launch_bounds__(BLOCK)
void cost_volume_kernel(const float* __restrict__ l,
                        const float* __restrict__ r,
                        float* __restrict__ out)
{
    __shared__ __align__(16) float sl[W_];
    __shared__ __align__(16) float sr[W_];

    const int row = blockIdx.x;        // row = bc*H + h,  bc = b*C + c
    const int h   = row % H_;
    const int bc  = row / H_;
    const int tid = threadIdx.x;

    const float* lrow = l + (size_t)row * W_;
    const float* rrow = r + (size_t)row * W_;

    // ---- Stage both rows into LDS via gfx1250 async global->LDS b128 copies.
    // Each active lane moves one 16-byte chunk; tracked by ASYNCcnt.
    if (tid < CHUNKS) {
        uint64_t ga = (uint64_t)(uintptr_t)(lrow + tid * 4);
        uint32_t la = (uint32_t)(uintptr_t)(&sl[tid * 4]);
        asm volatile("global_load_async_to_lds_b128 %0, %1, off"
                     :: "v"(la), "v"(ga) : "memory");
    } else if (tid < 2 * CHUNKS) {
        const int c4 = tid - CHUNKS;
        uint64_t ga = (uint64_t)(uintptr_t)(rrow + c4 * 4);
        uint32_t la = (uint32_t)(uintptr_t)(&sr[c4 * 4]);
        asm volatile("global_load_async_to_lds_b128 %0, %1, off"
                     :: "v"(la), "v"(ga) : "memory");
    }
#if __has_builtin(__builtin_amdgcn_s_wait_asynccnt)
    __builtin_amdgcn_s_wait_asynccnt(0);
#else
    asm volatile("s_wait_asynccnt 0" ::: "memory");
#endif
    __syncthreads();

    // ---- Emit all 48 disparities for this row from LDS.
    // out[((bc*D + d)*H + h)*W + w]
    float* obase = out + ((size_t)bc * MAX_DISP * H_ + h) * W_;

    const int total = MAX_DISP * CHUNKS;   // 3744 float4 stores per row
    for (int idx = tid; idx < total; idx += BLOCK) {
        const int d  = idx / CHUNKS;
        const int c4 = idx - d * CHUNKS;
        const int w0 = c4 * 4;

        v4f v;
        if (w0 >= d) {
            // all four lanes valid: 128-bit LDS read of l, scalar shifted r reads
            const v4f lv = *(const v4f*)(sl + w0);
            const int s = w0 - d;
            v.x = lv.x - sr[s + 0];
            v.y = lv.y - sr[s + 1];
            v.z = lv.z - sr[s + 2];
            v.w = lv.w - sr[s + 3];
        } else {
            float t[4];
#pragma unroll
            for (int j = 0; j < 4; ++j) {
                const int w = w0 + j;
                t[j] = (w >= d) ? (sl[w] - sr[w - d]) : PAD_VALUE;
            }
            v = (v4f){t[0], t[1], t[2], t[3]};
        }

        float* o = obase + (size_t)d * (H_ * W_) + w0;
        // Output is write-once: bypass cache retention with non-temporal b128 store.
        __builtin_nontemporal_store(v, (v4f*)o);
    }
}

extern "C" void kernel_launch(void* const* d_in, const int* in_sizes, int n_in,
                              void* d_out, int out_size, void* d_ws, size_t ws_size,
                              hipStream_t stream) {
    (void)in_sizes; (void)n_in; (void)out_size; (void)d_ws; (void)ws_size;
    const float* l = (const float*)d_in[0];
    const float* r = (const float*)d_in[1];
    float* out = (float*)d_out;

    dim3 grid(B_ * C_ * H_);   // 12288 workgroups, one (b,c,h) row each
    cost_volume_kernel<<<grid, BLOCK, 0, stream>>>(l, r, out);
}